// BaseEncoder_90400471646280
// MI455X (gfx1250) — compile-verified
//
#include <hip/hip_runtime.h>
#include <math.h>

typedef __attribute__((ext_vector_type(2))) float v2f;
typedef __attribute__((ext_vector_type(4))) float v4f;
typedef __attribute__((ext_vector_type(8))) float v8f;

#define BM 128
#define BN 64
#define BK 16
#define LDA_S 18          // A-tile LDS row stride (even -> 8B-aligned b64 fragment loads)
#define BP_STRIDE 65      // float2 per K-pair row in B tile (padded)

enum { F_EPI = 1, F_RELU2 = 2, F_BSCALE = 4 };

// Generic fp32 WMMA GEMM: C[M,N] = A[M,K] @ (diag(bscale) applied to B rows)[K,N]
// flags & F_EPI    : out = dvec[row]*acc + dvec[row]^2 * addsrc[row,col]  (GCN self-loop fold)
// flags & F_RELU2  : C2[row,col] = relu(out)
// flags & F_BSCALE : B row k scaled by bscale[k]
// Block tile 128x64, 8 waves (4 row-groups x 2 col-groups), each wave owns 32x32 C
// via 4 v8f accumulators; per K-step: 2 a-frag + 2 b-frag ds_load_b64 -> 4 WMMAs.
__global__ __launch_bounds__(256)
void wmma_gemm_f32(const float* __restrict__ A, const float* __restrict__ B,
                   const float* __restrict__ bscale,
                   float* __restrict__ C, float* __restrict__ C2,
                   const float* __restrict__ dvec, const float* __restrict__ addsrc,
                   int M, int N, int K, int lda, int ldb, int ldc, int flags)
{
    __shared__ float As[BM * LDA_S];                 // 9216 B
    __shared__ float Bs[(BK / 2) * BP_STRIDE * 2];   // 4160 B, K-paired layout

    const int m0   = blockIdx.y * BM;
    const int n0   = blockIdx.x * BN;
    const int tid  = threadIdx.x;
    const int wave = tid >> 5;
    const int lane = tid & 31;
    const int half = lane >> 4;          // 0: lanes 0-15 (K=k,k+1), 1: lanes 16-31 (K=k+2,k+3)
    const int l    = lane & 15;
    const int wrb  = (wave & 3) * 32;    // wave row base in block tile
    const int wcb  = (wave >> 2) * 32;   // wave col base in block tile

    v8f acc00 = {}, acc01 = {}, acc10 = {}, acc11 = {};

    // cooperative loader coordinates
    const int ar   = tid >> 1;           // 0..127 : A tile row
    const int acb  = (tid & 1) * 8;      // 0 or 8 : A tile col base (8 floats per thread)
    const int br   = tid >> 4;           // 0..15  : B tile row (K)
    const int bc   = (tid & 15) * 4;     // B tile col base (4 floats per thread)
    const int bp   = (br >> 1) * BP_STRIDE;  // K-pair index base
    const int bsub = br & 1;             // low/high element of the K pair

    // ---- staging helpers ----
    auto stage_a_fast = [&](int k0) __attribute__((always_inline)) {
        const float* ap = A + (long)(m0 + ar) * lda + k0 + acb;
        const v4f av0 = *(const v4f*)(ap);
        const v4f av1 = *(const v4f*)(ap + 4);
        float* as = &As[ar * LDA_S + acb];
        *(v2f*)(as + 0) = (v2f){av0.x, av0.y};
        *(v2f*)(as + 2) = (v2f){av0.z, av0.w};
        *(v2f*)(as + 4) = (v2f){av1.x, av1.y};
        *(v2f*)(as + 6) = (v2f){av1.z, av1.w};
        if (k0 + 2 * BK <= K) __builtin_prefetch(ap + BK, 0, 1);  // global_prefetch_b8
    };
    auto stage_a_guard = [&](int k0) __attribute__((always_inline)) {
        const int grow = m0 + ar;
#pragma unroll
        for (int i = 0; i < 8; ++i) {
            const int gk = k0 + acb + i;
            float v = 0.0f;
            if (grow < M && gk < K) v = A[(long)grow * lda + gk];
            As[ar * LDA_S + acb + i] = v;
        }
    };
    auto stage_b_fast = [&](int k0) __attribute__((always_inline)) {
        const int gk = k0 + br;
        const float s = (flags & F_BSCALE) ? bscale[gk] : 1.0f;
        const float* bpr = B + (long)gk * ldb + n0 + bc;
        const v4f bv = *(const v4f*)(bpr);
        Bs[(bp + bc + 0) * 2 + bsub] = bv.x * s;
        Bs[(bp + bc + 1) * 2 + bsub] = bv.y * s;
        Bs[(bp + bc + 2) * 2 + bsub] = bv.z * s;
        Bs[(bp + bc + 3) * 2 + bsub] = bv.w * s;
        if (k0 + 2 * BK <= K) __builtin_prefetch(bpr + (long)BK * ldb, 0, 1);
    };
    auto stage_b_guard = [&](int k0) __attribute__((always_inline)) {
        const int gk = k0 + br;
        float s = 1.0f;
        if ((flags & F_BSCALE) && gk < K) s = bscale[gk];
#pragma unroll
        for (int i = 0; i < 4; ++i) {
            const int gc = n0 + bc + i;
            float v = 0.0f;
            if (gk < K && gc < N) v = B[(long)gk * ldb + gc] * s;
            Bs[(bp + bc + i) * 2 + bsub] = v;
        }
    };
    // ---- 16x16x4 f32 WMMA over one staged K tile ----
    auto compute_tile = [&]() __attribute__((always_inline)) {
        __syncthreads();
#pragma unroll
        for (int kk = 0; kk < BK; kk += 4) {
            const int ka = kk + 2 * half;    // always even
            const v2f a0 = *(const v2f*)&As[(wrb + l) * LDA_S + ka];
            const v2f a1 = *(const v2f*)&As[(wrb + 16 + l) * LDA_S + ka];
            const v2f b0 = *(const v2f*)&Bs[((ka >> 1) * BP_STRIDE + wcb + l) * 2];
            const v2f b1 = *(const v2f*)&Bs[((ka >> 1) * BP_STRIDE + wcb + 16 + l) * 2];
            acc00 = __builtin_amdgcn_wmma_f32_16x16x4_f32(false, a0, false, b0,
                                                          (short)0, acc00, false, false);
            acc01 = __builtin_amdgcn_wmma_f32_16x16x4_f32(false, a0, false, b1,
                                                          (short)0, acc01, false, false);
            acc10 = __builtin_amdgcn_wmma_f32_16x16x4_f32(false, a1, false, b0,
                                                          (short)0, acc10, false, false);
            acc11 = __builtin_amdgcn_wmma_f32_16x16x4_f32(false, a1, false, b1,
                                                          (short)0, acc11, false, false);
        }
        __syncthreads();
    };

    const bool fullN = (n0 + BN <= N);
    const bool fullM = (m0 + BM <= M);
    const int  kFull = K & ~(BK - 1);    // K rounded down to tile multiple

    // hoisted specialization: predicate evaluated once, not per K-iteration
    if (fullM && fullN) {
        for (int k0 = 0; k0 < kFull; k0 += BK) {
            stage_a_fast(k0);
            stage_b_fast(k0);
            compute_tile();
        }
    } else if (fullM) {
        // N-tail column block: A staging can stay vectorized, only B needs guards
        for (int k0 = 0; k0 < kFull; k0 += BK) {
            stage_a_fast(k0);
            stage_b_guard(k0);
            compute_tile();
        }
    } else {
        for (int k0 = 0; k0 < kFull; k0 += BK) {
            stage_a_guard(k0);
            stage_b_guard(k0);
            compute_tile();
        }
    }
    if (kFull < K) {   // K-tail tile (zero-padded in LDS)
        stage_a_guard(kFull);
        stage_b_guard(kFull);
        compute_tile();
    }

    // ---- epilogue: C/D layout = VGPR v holds rows (base + 8*half + v), col = lane&15 ----
#pragma unroll
    for (int r = 0; r < 2; ++r) {
#pragma unroll
        for (int v = 0; v < 8; ++v) {
            const int row = m0 + wrb + r * 16 + half * 8 + v;
            if (row >= M) continue;
            const float dd = (flags & F_EPI) ? dvec[row] : 0.0f;
#pragma unroll
            for (int j = 0; j < 2; ++j) {
                const int col = n0 + wcb + j * 16 + l;
                if (col >= N) continue;
                float o = r ? (j ? acc11[v] : acc10[v]) : (j ? acc01[v] : acc00[v]);
                if (flags & F_EPI)
                    o = dd * o + dd * dd * addsrc[(long)row * ldc + col];
                C[(long)row * ldc + col] = o;
                if (flags & F_RELU2)
                    C2[(long)row * ldc + col] = fmaxf(o, 0.0f);
            }
        }
    }
}

// out[row] = sum of row (one 256-thread block per row)
__global__ __launch_bounds__(256)
void rowsum_kernel(const float* __restrict__ A, float* __restrict__ out, int M, int N)
{
    const int row = blockIdx.x;
    const float* p = A + (long)row * N;
    float s = 0.0f;
    for (int c = threadIdx.x; c < N; c += 256) s += p[c];
    for (int off = 16; off > 0; off >>= 1) s += __shfl_down(s, off, 32);
    __shared__ float red[8];
    if ((threadIdx.x & 31) == 0) red[threadIdx.x >> 5] = s;
    __syncthreads();
    if (threadIdx.x == 0) {
        float t = 0.0f;
        for (int w = 0; w < 8; ++w) t += red[w];
        out[row] = t;
    }
}

__global__ __launch_bounds__(256)
void rsqrt_deg_kernel(const float* __restrict__ adjsum, float* __restrict__ d, int n)
{
    int i = blockIdx.x * 256 + threadIdx.x;
    if (i < n) d[i] = rsqrtf(adjsum[i] + 1.0f);  // self-loop adds 1; deg >= 1
}

// g = sigmoid( L2normalize( vsum/gnsum ) ), one 64-thread block per row
__global__ __launch_bounds__(64)
void readout_kernel(const float* __restrict__ vsum, const float* __restrict__ gnsum,
                    float* __restrict__ g, int D)
{
    const int row = blockIdx.x;
    const int t = threadIdx.x;
    const float val = vsum[(long)row * D + t] / gnsum[row];
    __shared__ float red[64];
    red[t] = val * val;
    __syncthreads();
    for (int off = 32; off > 0; off >>= 1) {
        if (t < off) red[t] += red[t + off];
        __syncthreads();
    }
    const float nrm = fmaxf(sqrtf(red[0]), 1e-12f);
    const float x = val / nrm;
    g[(long)row * D + t] = 1.0f / (1.0f + expf(-x));
}

// Bilinear discriminator: ret[i] = [emb_i W g_i, emb_a_i W g_i] + b
//                         ret_a[i] = [emb_a_i W g_a_i, emb_i W g_a_i] + b
__global__ __launch_bounds__(64)
void disc_kernel(const float* __restrict__ emb, const float* __restrict__ emb_a,
                 const float* __restrict__ g, const float* __restrict__ g_a,
                 const float* __restrict__ W, const float* __restrict__ bptr,
                 float* __restrict__ ret, float* __restrict__ ret_a, int D)
{
    const int row = blockIdx.x;
    const int e = threadIdx.x;
    __shared__ float hp[64], hm[64], red[64];
    hp[e] = emb[(long)row * D + e];
    hm[e] = emb_a[(long)row * D + e];
    __syncthreads();
    float tpl = 0.0f, tmi = 0.0f;
    for (int k = 0; k < 64; ++k) {
        const float w = W[k * 64 + e];
        tpl += hp[k] * w;
        tmi += hm[k] * w;
    }
    const float ge  = g[(long)row * D + e];
    const float gae = g_a[(long)row * D + e];
    const float b = bptr[0];
    float vals[4] = { tpl * ge, tmi * ge, tmi * gae, tpl * gae };
    float sums[4];
#pragma unroll
    for (int q = 0; q < 4; ++q) {
        red[e] = vals[q];
        __syncthreads();
        for (int off = 32; off > 0; off >>= 1) {
            if (e < off) red[e] += red[e + off];
            __syncthreads();
        }
        sums[q] = red[0];
        __syncthreads();
    }
    if (e == 0) {
        ret[row * 2 + 0]   = sums[0] + b;
        ret[row * 2 + 1]   = sums[1] + b;
        ret_a[row * 2 + 0] = sums[2] + b;
        ret_a[row * 2 + 1] = sums[3] + b;
    }
}

extern "C" void kernel_launch(void* const* d_in, const int* in_sizes, int n_in,
                              void* d_out, int out_size, void* d_ws, size_t ws_size,
                              hipStream_t stream)
{
    (void)in_sizes; (void)n_in; (void)out_size; (void)ws_size;

    const float* feat   = (const float*)d_in[0];
    const float* feat_a = (const float*)d_in[1];
    const float* adj    = (const float*)d_in[2];
    const float* gneigh = (const float*)d_in[3];
    const float* W1     = (const float*)d_in[4];
    const float* W2     = (const float*)d_in[5];
    const float* discW  = (const float*)d_in[6];
    const float* discb  = (const float*)d_in[7];

    const int N = 8192, FIN = 3000, FOUT = 64;

    // outputs: (z, h, ret, ret_a, emb, emb_a) concatenated flat
    float* out      = (float*)d_out;
    float* z_out    = out;                          // [N,64]
    float* h_out    = z_out + (long)N * FOUT;       // [N,3000]
    float* ret_out  = h_out + (long)N * FIN;        // [N,2]
    float* reta_out = ret_out + (long)N * 2;        // [N,2]
    float* emb_out  = reta_out + (long)N * 2;       // [N,64]
    float* emba_out = emb_out + (long)N * FOUT;     // [N,64]

    // workspace layout
    float* ws = (float*)d_ws;
    float* dvec   = ws;  ws += N;
    float* adjsum = ws;  ws += N;
    float* gnsum  = ws;  ws += N;
    float* Y      = ws;  ws += (long)N * FOUT;
    float* Ya     = ws;  ws += (long)N * FOUT;
    float* ztmp   = ws;  ws += (long)N * FOUT;
    float* vsum   = ws;  ws += (long)N * FOUT;
    float* vsum_a = ws;  ws += (long)N * FOUT;
    float* g      = ws;  ws += (long)N * FOUT;
    float* g_a    = ws;  ws += (long)N * FOUT;
    float* U      = ws;  ws += (long)N * FIN;       // z @ W2, [N,3000]

    const dim3 blk(256);

    // 1) degree/rowsum vectors (adj symmetric -> colsum == rowsum)
    rowsum_kernel<<<N, 256, 0, stream>>>(adj, adjsum, N, N);
    rowsum_kernel<<<N, 256, 0, stream>>>(gneigh, gnsum, N, N);
    rsqrt_deg_kernel<<<(N + 255) / 256, 256, 0, stream>>>(adjsum, dvec, N);

    // 2) Y = feat @ W1 ; Ya = feat_a @ W1
    {
        dim3 grid((FOUT + BN - 1) / BN, N / BM);
        wmma_gemm_f32<<<grid, blk, 0, stream>>>(feat, W1, nullptr, Y, nullptr, nullptr, nullptr,
                                                N, FOUT, FIN, FIN, FOUT, FOUT, 0);
        wmma_gemm_f32<<<grid, blk, 0, stream>>>(feat_a, W1, nullptr, Ya, nullptr, nullptr, nullptr,
                                                N, FOUT, FIN, FIN, FOUT, FOUT, 0);
    }

    // 3) z = d ∘ (adj @ (d ∘ Y)) + d^2 ∘ Y ; emb = relu(z)  (fused epilogue)
    {
        dim3 grid((FOUT + BN - 1) / BN, N / BM);
        wmma_gemm_f32<<<grid, blk, 0, stream>>>(adj, Y, dvec, z_out, emb_out, dvec, Y,
                                                N, FOUT, N, N, FOUT, FOUT,
                                                F_EPI | F_RELU2 | F_BSCALE);
        wmma_gemm_f32<<<grid, blk, 0, stream>>>(adj, Ya, dvec, ztmp, emba_out, dvec, Ya,
                                                N, FOUT, N, N, FOUT, FOUT,
                                                F_EPI | F_RELU2 | F_BSCALE);
    }

    // 4) vsum = graph_neigh @ emb ; vsum_a = graph_neigh @ emb_a
    {
        dim3 grid((FOUT + BN - 1) / BN, N / BM);
        wmma_gemm_f32<<<grid, blk, 0, stream>>>(gneigh, emb_out, nullptr, vsum, nullptr, nullptr,
                                                nullptr, N, FOUT, N, N, FOUT, FOUT, 0);
        wmma_gemm_f32<<<grid, blk, 0, stream>>>(gneigh, emba_out, nullptr, vsum_a, nullptr, nullptr,
                                                nullptr, N, FOUT, N, N, FOUT, FOUT, 0);
    }

    // 5) readout -> g, g_a
    readout_kernel<<<N, 64, 0, stream>>>(vsum, gnsum, g, FOUT);
    readout_kernel<<<N, 64, 0, stream>>>(vsum_a, gnsum, g_a, FOUT);

    // 6) discriminator -> ret, ret_a
    disc_kernel<<<N, 64, 0, stream>>>(emb_out, emba_out, g, g_a, discW, discb,
                                      ret_out, reta_out, FOUT);

    // 7) U = z @ W2
    {
        dim3 grid((FIN + BN - 1) / BN, N / BM);
        wmma_gemm_f32<<<grid, blk, 0, stream>>>(z_out, W2, nullptr, U, nullptr, nullptr, nullptr,
                                                N, FIN, FOUT, FOUT, FIN, FIN, 0);
    }

    // 8) h = d ∘ (adj @ (d ∘ U)) + d^2 ∘ U   (dominant 403-GFLOP GEMM, fused epilogue)
    {
        dim3 grid((FIN + BN - 1) / BN, N / BM);
        wmma_gemm_f32<<<grid, blk, 0, stream>>>(adj, U, dvec, h_out, nullptr, dvec, U,
                                                N, FIN, N, N, FIN, FIN,
                                                F_EPI | F_BSCALE);
    }
}